// SelfAttention_23699629539656
// MI455X (gfx1250) — compile-verified
//
#include <hip/hip_runtime.h>
#include <math.h>

// ---------------------------------------------------------------------------
// CDNA5 (gfx1250) fp32 WMMA self-attention.
// D(16x16,f32) = A(16x4,f32) x B(4x16,f32) + C   via v_wmma_f32_16x16x4_f32
// Tile staging uses gfx1250 async LDS copies (global_load_async_to_lds_b128,
// ASYNCcnt) so copies bypass VGPRs and overlap with WMMA compute.
// GEMMs use 128x64 block tiles: each wave computes a 2x2 grid of 16x16 tiles,
// giving 4 WMMAs per 4 LDS fragment loads (2x the FLOPs per LDS byte of the
// previous 2-tile version).
//
// Fragment layouts (wave32), per CDNA5 ISA 7.12.2:
//   A 16x4 : lane L (0-15): M=L, VGPR0/1 = K0/K1 ; lanes 16-31: K2/K3
//   B 4x16 : lane L: N=L&15, VGPR0/1 = K{2*(L>>4)} / K{2*(L>>4)+1}
//   C/D    : lane L: N=L&15, VGPR i = row (i + 8*(L>>4))
// ---------------------------------------------------------------------------

typedef __attribute__((ext_vector_type(2))) float v2f;
typedef __attribute__((ext_vector_type(8))) float v8f;

#define WMMA_F32(a, b, c) \
    __builtin_amdgcn_wmma_f32_16x16x4_f32(false, (a), false, (b), (short)0, (c), false, false)

#define CB 2
#define CN 2048
#define CC 2048
#define CH 16
#define CD 128
#define CM (CB * CN)     /* 4096 */
#define CFQKV (3 * CC)   /* 6144 */
#define ATT_SCALE 0.08838834764831845f /* 128^-0.5 */
#define RMS_EPS 1e-5f

// Async copy 16B from global to LDS; tracked by ASYNCcnt, no VGPR round-trip.
__device__ __forceinline__ void async_copy_b128(unsigned lds_off, const float* gaddr)
{
    asm volatile("global_load_async_to_lds_b128 %0, %1, off"
                 :: "v"(lds_off), "v"(gaddr) : "memory");
}
__device__ __forceinline__ void wait_async()
{
    asm volatile("s_wait_asynccnt 0x0" ::: "memory");
}
__device__ __forceinline__ unsigned lds_addr(const void* p)
{
    return (unsigned)(uintptr_t)p;   // LDS aperture: low 32 bits = LDS offset
}

// ---------------------------------------------------------------------------
// Kernel 1: QKV GEMM.  Y[m,f] = sum_k X[m,k] * Wq[f,k]; scatter into
// q/k/v buffers laid out [B][H][N][D].
// 128x64 block tile / 256 threads; k staged 32 at a time, double-buffered.
// Wave w: m-tiles {w>>1, (w>>1)+4}, n-tiles {2*(w&1), 2*(w&1)+1}.
// ---------------------------------------------------------------------------
__global__ __launch_bounds__(256) void qkv_gemm_kernel(
    const float* __restrict__ X, const float* __restrict__ W,
    float* __restrict__ qb, float* __restrict__ kb, float* __restrict__ vb)
{
    // 36-word row stride: 144 B (16B aligned) and conflict-free frag reads.
    __shared__ float Xs[2][128 * 36];
    __shared__ float Ws[2][64 * 36];

    const int f0 = blockIdx.x * 64;
    const int m0 = blockIdx.y * 128;
    const int t = threadIdx.x;
    const int lane = t & 31;
    const int wv = t >> 5;
    const int hl = lane >> 4;
    const int ln = lane & 15;
    const int wvm = wv >> 1;   // 0..3 -> m-tiles wvm and wvm+4
    const int wvn = wv & 1;    // 0..1 -> n-tiles 2*wvn and 2*wvn+1

    auto stage = [&](int bb, int k0) {
#pragma unroll
        for (int i = 0; i < 4; ++i) {
            int slot = t + 256 * i;        // 1024 float4 slots (128 rows x 8)
            int row = slot >> 3;
            int c4 = (slot & 7) * 4;
            async_copy_b128(lds_addr(&Xs[bb][row * 36 + c4]),
                            &X[(size_t)(m0 + row) * CC + k0 + c4]);
        }
#pragma unroll
        for (int i = 0; i < 2; ++i) {
            int slot = t + 256 * i;        // 512 float4 slots (64 rows x 8)
            int row = slot >> 3;
            int c4 = (slot & 7) * 4;
            async_copy_b128(lds_addr(&Ws[bb][row * 36 + c4]),
                            &W[(size_t)(f0 + row) * CC + k0 + c4]);
        }
    };

    v8f accs[2][2];
#pragma unroll
    for (int mi = 0; mi < 2; ++mi)
#pragma unroll
        for (int ni = 0; ni < 2; ++ni) accs[mi][ni] = (v8f){};

    stage(0, 0);
    wait_async();
    __syncthreads();

    for (int k0 = 0; k0 < CC; k0 += 32) {
        const int cur = (k0 >> 5) & 1;
        if (k0 + 32 < CC) stage(cur ^ 1, k0 + 32);   // prefetch next stage

        const float* Xb = &Xs[cur][0];
        const float* Wb = &Ws[cur][0];
#pragma unroll
        for (int s = 0; s < 8; ++s) {
            v2f a0 = *(const v2f*)&Xb[(wvm * 16 + ln) * 36 + 4 * s + 2 * hl];
            v2f a1 = *(const v2f*)&Xb[((wvm + 4) * 16 + ln) * 36 + 4 * s + 2 * hl];
            v2f b0 = *(const v2f*)&Wb[((wvn * 2) * 16 + ln) * 36 + 4 * s + 2 * hl];
            v2f b1 = *(const v2f*)&Wb[((wvn * 2 + 1) * 16 + ln) * 36 + 4 * s + 2 * hl];
            accs[0][0] = WMMA_F32(a0, b0, accs[0][0]);
            accs[0][1] = WMMA_F32(a0, b1, accs[0][1]);
            accs[1][0] = WMMA_F32(a1, b0, accs[1][0]);
            accs[1][1] = WMMA_F32(a1, b1, accs[1][1]);
        }
        wait_async();
        __syncthreads();
    }

    // Scatter: f -> (which, h, d); tile columns never straddle a head (16 | 128).
#pragma unroll
    for (int ni = 0; ni < 2; ++ni) {
        const int f = f0 + (wvn * 2 + ni) * 16 + ln;
        const int which = f >> 11;
        const int fr = f & 2047;
        const int h = fr >> 7;
        const int d = fr & 127;
        float* dst = (which == 0) ? qb : ((which == 1) ? kb : vb);
#pragma unroll
        for (int mi = 0; mi < 2; ++mi) {
            const int mrow0 = m0 + (wvm + 4 * mi) * 16 + 8 * hl;
#pragma unroll
            for (int i = 0; i < 8; ++i) {
                int m = mrow0 + i;
                int b = m >> 11;
                int n = m & 2047;
                dst[((size_t)(b * CH + h) * CN + n) * CD + d] = accs[mi][ni][i];
            }
        }
    }
}

// ---------------------------------------------------------------------------
// Kernel 2: per-(b,n) sum of squares over C=2048 -> inverse RMS.
// grid: (B*N, 2)   y==0 -> q, y==1 -> k
// ---------------------------------------------------------------------------
__global__ __launch_bounds__(256) void rms_reduce_kernel(
    const float* __restrict__ qb, const float* __restrict__ kb,
    float* __restrict__ rq, float* __restrict__ rk)
{
    __shared__ float red[8];
    const int bn = blockIdx.x;
    const int b = bn >> 11;
    const int n = bn & 2047;
    const float* buf = blockIdx.y ? kb : qb;

    float ssq = 0.f;
#pragma unroll
    for (int i = 0; i < 8; ++i) {
        int j = threadIdx.x + 256 * i;   // channel c = h*128+d
        int h = j >> 7;
        int d = j & 127;
        float v = buf[((size_t)(b * CH + h) * CN + n) * CD + d];
        ssq += v * v;
    }
#pragma unroll
    for (int m = 16; m >= 1; m >>= 1) ssq += __shfl_xor(ssq, m, 32);
    if ((threadIdx.x & 31) == 0) red[threadIdx.x >> 5] = ssq;
    __syncthreads();
    if (threadIdx.x == 0) {
        float s = 0.f;
#pragma unroll
        for (int i = 0; i < 8; ++i) s += red[i];
        float inv = rsqrtf(s * (1.0f / (float)CC) + RMS_EPS);
        (blockIdx.y ? rk : rq)[bn] = inv;
    }
}

// ---------------------------------------------------------------------------
// Kernel 3: RMS-norm scale * norm-weight, interleaved RoPE, and fold the
// attention scale D^-0.5 into q.  In-place on q/k buffers.
// grid: (B*H*N*D/2 / 256, 2)
// ---------------------------------------------------------------------------
__global__ __launch_bounds__(256) void rope_kernel(
    float* __restrict__ qb, float* __restrict__ kb,
    const float* __restrict__ rq, const float* __restrict__ rk,
    const float* __restrict__ qw, const float* __restrict__ kw,
    const float* __restrict__ cosb, const float* __restrict__ sinb)
{
    const int isK = blockIdx.y;
    float* buf = isK ? kb : qb;
    const float* rms = isK ? rk : rq;
    const float* wn = isK ? kw : qw;
    const float outscale = isK ? 1.0f : ATT_SCALE;

    int p = blockIdx.x * 256 + threadIdx.x;  // pair index
    int d = (p & 63) * 2;
    int n = (p >> 6) & 2047;
    int bh = p >> 17;                        // b*H + h
    int b = bh >> 4;
    int h = bh & 15;

    size_t base = ((size_t)bh * CN + n) * CD + d;
    float r = rms[b * CN + n];
    float v0 = buf[base] * r * wn[h * CD + d];
    float v1 = buf[base + 1] * r * wn[h * CD + d + 1];
    float c0 = cosb[n * CD + d], c1 = cosb[n * CD + d + 1];
    float s0 = sinb[n * CD + d], s1 = sinb[n * CD + d + 1];
    buf[base]     = (v0 * c0 - v1 * s0) * outscale;
    buf[base + 1] = (v1 * c1 + v0 * s1) * outscale;
}

// ---------------------------------------------------------------------------
// Kernel 4: flash attention.  Block = 8 waves; each wave privately owns 16
// query rows (no cross-wave softmax state).  KV tile = 32, async-staged LDS.
// grid: (N/128, B*H)
// ---------------------------------------------------------------------------
__global__ __launch_bounds__(256) void flash_attn_kernel(
    const float* __restrict__ qb, const float* __restrict__ kb,
    const float* __restrict__ vb, float* __restrict__ ob)
{
    __shared__ float Ks[32 * 132];      // [kvrow][d], padded (528 B rows)
    __shared__ float Vs[32 * 132];
    __shared__ float Ps[8 * 16 * 36];   // per-wave P scratch [16][36]

    const int t = threadIdx.x;
    const int lane = t & 31;
    const int wv = t >> 5;
    const int hl = lane >> 4;
    const int ln = lane & 15;
    const int bh = blockIdx.y;          // b*H + h
    const int b = bh >> 4;
    const int h = bh & 15;
    const int q0 = blockIdx.x * 128 + wv * 16;

    const float* Qg = qb + (size_t)bh * CN * CD;
    const float* Kg = kb + (size_t)bh * CN * CD;
    const float* Vg = vb + (size_t)bh * CN * CD;

    // Q fragments for this wave's 16 rows (already scaled by D^-0.5).
    v2f qa[32];
#pragma unroll
    for (int s = 0; s < 32; ++s)
        qa[s] = *(const v2f*)&Qg[(size_t)(q0 + ln) * CD + 4 * s + 2 * hl];

    v8f oacc[8];
#pragma unroll
    for (int j = 0; j < 8; ++j) oacc[j] = (v8f){};
    float mstat[8], lstat[8];
#pragma unroll
    for (int i = 0; i < 8; ++i) { mstat[i] = -INFINITY; lstat[i] = 0.f; }

    float* Pw = &Ps[wv * 16 * 36];

    for (int kv0 = 0; kv0 < CN; kv0 += 32) {
        // Async-stage K,V tiles (32x128 each): 8 x b128 per thread in flight.
#pragma unroll
        for (int i = 0; i < 4; ++i) {
            int slot = t + 256 * i;     // 1024 float4 slots (32 rows x 32)
            int row = slot >> 5;
            int c4 = (slot & 31) * 4;
            async_copy_b128(lds_addr(&Ks[row * 132 + c4]),
                            &Kg[(size_t)(kv0 + row) * CD + c4]);
            async_copy_b128(lds_addr(&Vs[row * 132 + c4]),
                            &Vg[(size_t)(kv0 + row) * CD + c4]);
        }
        wait_async();
        __syncthreads();

        // S tile 16x32 = two 16x16 WMMA tiles, K-dim = D = 128.
        v8f ct0 = {};
        v8f ct1 = {};
#pragma unroll
        for (int s = 0; s < 32; ++s) {
            v2f b0 = *(const v2f*)&Ks[ln * 132 + 4 * s + 2 * hl];
            v2f b1 = *(const v2f*)&Ks[(16 + ln) * 132 + 4 * s + 2 * hl];
            ct0 = WMMA_F32(qa[s], b0, ct0);
            ct1 = WMMA_F32(qa[s], b1, ct1);
        }

        // Online softmax over this wave's 16 rows (row i+8*hl lives in elem i).
#pragma unroll
        for (int i = 0; i < 8; ++i) {
            float vmax = fmaxf(ct0[i], ct1[i]);
            vmax = fmaxf(vmax, __shfl_xor(vmax, 1, 32));
            vmax = fmaxf(vmax, __shfl_xor(vmax, 2, 32));
            vmax = fmaxf(vmax, __shfl_xor(vmax, 4, 32));
            vmax = fmaxf(vmax, __shfl_xor(vmax, 8, 32));
            float newm = fmaxf(mstat[i], vmax);
            float fac = __expf(mstat[i] - newm);
            float p0 = __expf(ct0[i] - newm);
            float p1 = __expf(ct1[i] - newm);
            float rs = p0 + p1;
            rs += __shfl_xor(rs, 1, 32);
            rs += __shfl_xor(rs, 2, 32);
            rs += __shfl_xor(rs, 4, 32);
            rs += __shfl_xor(rs, 8, 32);
            lstat[i] = lstat[i] * fac + rs;
            mstat[i] = newm;
            int prow = i + 8 * hl;
            Pw[prow * 36 + ln] = p0;
            Pw[prow * 36 + 16 + ln] = p1;
#pragma unroll
            for (int j = 0; j < 8; ++j) oacc[j][i] *= fac;
        }

        // O += P (16x32) x V (32x128); A-frags re-read from per-wave LDS.
        v2f ap[8];
#pragma unroll
        for (int s = 0; s < 8; ++s)
            ap[s] = *(const v2f*)&Pw[ln * 36 + 4 * s + 2 * hl];
#pragma unroll
        for (int j = 0; j < 8; ++j) {
#pragma unroll
            for (int s = 0; s < 8; ++s) {
                v2f bv;
                bv.x = Vs[(4 * s + 2 * hl) * 132 + j * 16 + ln];
                bv.y = Vs[(4 * s + 2 * hl + 1) * 132 + j * 16 + ln];
                oacc[j] = WMMA_F32(ap[s], bv, oacc[j]);
            }
        }
        __syncthreads();   // K/V (and P) consumed; safe to overwrite next iter
    }

    // Normalize and store to [B][N][C] (c = h*128 + col).
#pragma unroll
    for (int i = 0; i < 8; ++i) {
        int row = q0 + i + 8 * hl;
        float inv = 1.0f / lstat[i];
#pragma unroll
        for (int j = 0; j < 8; ++j)
            ob[((size_t)(b * CN + row)) * CC + h * CD + j * 16 + ln] = oacc[j][i] * inv;
    }
}

// ---------------------------------------------------------------------------
// Kernel 5: output projection + bias.  Y[m,o] = sum_c Ob[m,c]*Wp[o,c] + bias[o]
// Same 128x64 double-buffered async-staged mainloop as kernel 1.
// ---------------------------------------------------------------------------
__global__ __launch_bounds__(256) void proj_gemm_kernel(
    const float* __restrict__ X, const float* __restrict__ W,
    const float* __restrict__ bias, float* __restrict__ Y)
{
    __shared__ float Xs[2][128 * 36];
    __shared__ float Ws[2][64 * 36];

    const int f0 = blockIdx.x * 64;
    const int m0 = blockIdx.y * 128;
    const int t = threadIdx.x;
    const int lane = t & 31;
    const int wv = t >> 5;
    const int hl = lane >> 4;
    const int ln = lane & 15;
    const int wvm = wv >> 1;
    const int wvn = wv & 1;

    auto stage = [&](int bb, int k0) {
#pragma unroll
        for (int i = 0; i < 4; ++i) {
            int slot = t + 256 * i;
            int row = slot >> 3;
            int c4 = (slot & 7) * 4;
            async_copy_b128(lds_addr(&Xs[bb][row * 36 + c4]),
                            &X[(size_t)(m0 + row) * CC + k0 + c4]);
        }
#pragma unroll
        for (int i = 0; i < 2; ++i) {
            int slot = t + 256 * i;
            int row = slot >> 3;
            int c4 = (slot & 7) * 4;
            async_copy_b128(lds_addr(&Ws[bb][row * 36 + c4]),
                            &W[(size_t)(f0 + row) * CC + k0 + c4]);
        }
    };

    v8f accs[2][2];
#pragma unroll
    for (int mi = 0; mi < 2; ++mi)
#pragma unroll
        for (int ni = 0; ni < 2; ++ni) accs[mi][ni] = (v8f){};

    stage(0, 0);
    wait_async();
    __syncthreads();

    for (int k0 = 0; k0 < CC; k0 += 32) {
        const int cur = (k0 >> 5) & 1;
        if (k0 + 32 < CC) stage(cur ^ 1, k0 + 32);

        const float* Xb = &Xs[cur][0];
        const float* Wb = &Ws[cur][0];
#pragma unroll
        for (int s = 0; s < 8; ++s) {
            v2f a0 = *(const v2f*)&Xb[(wvm * 16 + ln) * 36 + 4 * s + 2 * hl];
            v2f a1 = *(const v2f*)&Xb[((wvm + 4) * 16 + ln) * 36 + 4 * s + 2 * hl];
            v2f b0 = *(const v2f*)&Wb[((wvn * 2) * 16 + ln) * 36 + 4 * s + 2 * hl];
            v2f b1 = *(const v2f*)&Wb[((wvn * 2 + 1) * 16 + ln) * 36 + 4 * s + 2 * hl];
            accs[0][0] = WMMA_F32(a0, b0, accs[0][0]);
            accs[0][1] = WMMA_F32(a0, b1, accs[0][1]);
            accs[1][0] = WMMA_F32(a1, b0, accs[1][0]);
            accs[1][1] = WMMA_F32(a1, b1, accs[1][1]);
        }
        wait_async();
        __syncthreads();
    }

#pragma unroll
    for (int ni = 0; ni < 2; ++ni) {
        const int f = f0 + (wvn * 2 + ni) * 16 + ln;
        const float bvv = bias[f];
#pragma unroll
        for (int mi = 0; mi < 2; ++mi) {
            const int mrow0 = m0 + (wvm + 4 * mi) * 16 + 8 * hl;
#pragma unroll
            for (int i = 0; i < 8; ++i)
                Y[(size_t)(mrow0 + i) * CC + f] = accs[mi][ni][i] + bvv;
        }
    }
}

// ---------------------------------------------------------------------------
extern "C" void kernel_launch(void* const* d_in, const int* in_sizes, int n_in,
                              void* d_out, int out_size, void* d_ws, size_t ws_size,
                              hipStream_t stream)
{
    const float* x      = (const float*)d_in[0];
    const float* w_qkv  = (const float*)d_in[1];
    const float* qw     = (const float*)d_in[2];
    const float* kw     = (const float*)d_in[3];
    const float* cosb   = (const float*)d_in[4];
    const float* sinb   = (const float*)d_in[5];
    const float* w_proj = (const float*)d_in[6];
    const float* b_proj = (const float*)d_in[7];
    float* out = (float*)d_out;

    // Workspace carve-up (~128 MB of fp32).
    const size_t per_head = (size_t)CB * CH * CN * CD;  // 8388608
    float* ws = (float*)d_ws;
    float* qb = ws;
    float* kb = qb + per_head;
    float* vb = kb + per_head;
    float* ob = vb + per_head;           // [B][N][C]
    float* rq = ob + (size_t)CB * CN * CC;
    float* rk = rq + (size_t)CB * CN;

    qkv_gemm_kernel<<<dim3(CFQKV / 64, CM / 128), 256, 0, stream>>>(x, w_qkv, qb, kb, vb);
    rms_reduce_kernel<<<dim3(CB * CN, 2), 256, 0, stream>>>(qb, kb, rq, rk);
    rope_kernel<<<dim3((CB * CH * CN * CD / 2) / 256, 2), 256, 0, stream>>>(
        qb, kb, rq, rk, qw, kw, cosb, sinb);
    flash_attn_kernel<<<dim3(CN / 128, CB * CH), 256, 0, stream>>>(qb, kb, vb, ob);
    proj_gemm_kernel<<<dim3(CC / 64, CM / 128), 256, 0, stream>>>(ob, w_proj, b_proj, out);
}